// MoNet_54511724921558
// MI455X (gfx1250) — compile-verified
//
#include <hip/hip_runtime.h>
#include <math.h>

typedef __attribute__((ext_vector_type(2))) float v2f;
typedef __attribute__((ext_vector_type(8))) float v8f;

#define N_   100000
#define E_   1600000
#define N1_  50000
#define E1_  800000
#define N2_  25000
#define E2_  400000
#define B_   128
#define EPS_ 1e-15f
#define ENC_NEGINF 0x007FFFFFu   // monotonic-uint encoding of -inf

static __device__ __forceinline__ float eluf(float x) {
    return x > 0.0f ? x : (__expf(x) - 1.0f);
}

static __device__ __forceinline__ unsigned enc_f32(float f) {
    unsigned b = __float_as_uint(f);
    return (b & 0x80000000u) ? ~b : (b | 0x80000000u);
}
static __device__ __forceinline__ float dec_f32(unsigned u) {
    return (u & 0x80000000u) ? __uint_as_float(u ^ 0x80000000u)
                             : __uint_as_float(~u);
}

// ---------------- generic fp32 WMMA GEMM: C[M,Nc] = A[M,KD] @ B[KD,Nc] ----
// row-major A/B/C, Nc multiple of 16, KD multiple of 4. One 16x16 tile/wave.
template <int KD>
__global__ void wmma_gemm(const float* __restrict__ A, const float* __restrict__ B,
                          float* __restrict__ C, int M, int Nc)
{
    int wave = (int)((blockIdx.x * (long)blockDim.x + threadIdx.x) >> 5);
    int lane = threadIdx.x & 31;
    int nTn  = Nc >> 4;
    int nTm  = (M + 15) >> 4;
    if (wave >= nTm * nTn) return;            // uniform per wave: EXEC stays all-1s
    int tm = wave / nTn, tn = wave % nTn;

    int l16 = lane & 15;
    int kh  = lane >> 4;                      // half-wave selects K pair

    int arow = tm * 16 + l16;
    if (arow >= M) arow = M - 1;              // clamp; garbage rows never stored
    const float* Ap = A + (size_t)arow * KD + 2 * kh;
    const float* Bp = B + (size_t)(2 * kh) * Nc + tn * 16 + l16;

    v8f c = {};
#pragma unroll
    for (int k = 0; k < KD; k += 4) {
        v2f a, b;
        a.x = Ap[k + 0];
        a.y = Ap[k + 1];
        b.x = Bp[(size_t)(k + 0) * Nc];
        b.y = Bp[(size_t)(k + 1) * Nc];
        c = __builtin_amdgcn_wmma_f32_16x16x4_f32(false, a, false, b,
                                                  (short)0, c, false, false);
    }

    int ncol = tn * 16 + l16;
#pragma unroll
    for (int i = 0; i < 8; i++) {
        int r = tm * 16 + i + 8 * kh;         // C/D layout: vgpr i -> row i (+8 hi half)
        if (r < M) C[(size_t)r * Nc + ncol] = c[i];
    }
}

// ---------------- layer-1 node transform: y1[n][j] = x[n] * g1[j], j<96 ----
__global__ void l1_node(const float* __restrict__ x, const float* __restrict__ g1,
                        float* __restrict__ y1)
{
    long idx = blockIdx.x * (long)blockDim.x + threadIdx.x;
    if (idx >= (long)N_ * 96) return;
    int n = (int)(idx / 96), j = (int)(idx % 96);
    y1[idx] = x[n] * g1[j];
}

// ---------------- Gaussian mixture weights ------------------------------
static __device__ __forceinline__ void gauss3(float p0, float p1,
                                              const float* __restrict__ mu,
                                              const float* __restrict__ sig,
                                              float* __restrict__ out)
{
#pragma unroll
    for (int k = 0; k < 3; k++) {
        float d0 = p0 - mu[2 * k], d1 = p1 - mu[2 * k + 1];
        float s0 = sig[2 * k],     s1 = sig[2 * k + 1];
        float t = -0.5f * (d0 * d0 / (EPS_ + s0 * s0) + d1 * d1 / (EPS_ + s1 * s1));
        out[k] = __expf(t);
    }
}

__global__ void gauss_from_attr(const float* __restrict__ pseudo,
                                const float* __restrict__ mu, const float* __restrict__ sig,
                                float* __restrict__ gauss, int E)
{
    int e = blockIdx.x * blockDim.x + threadIdx.x;
    if (e >= E) return;
    gauss3(pseudo[2 * (size_t)e], pseudo[2 * (size_t)e + 1], mu, sig,
           &gauss[3 * (size_t)e]);
}

__global__ void cart_absmax(const int* __restrict__ ei, const float* __restrict__ pos,
                            float* __restrict__ mx, int E)
{
    int e = blockIdx.x * blockDim.x + threadIdx.x;
    float v = 0.0f;
    if (e < E) {
        int r = ei[e], c = ei[E + e];
        float c0 = pos[2 * r]     - pos[2 * c];
        float c1 = pos[2 * r + 1] - pos[2 * c + 1];
        v = fmaxf(fabsf(c0), fabsf(c1));
    }
    for (int off = 16; off > 0; off >>= 1)      // wave32 reduction
        v = fmaxf(v, __shfl_xor(v, off));
    if ((threadIdx.x & 31) == 0)
        atomicMax((unsigned*)mx, __float_as_uint(v));   // non-negative: bit order OK
}

__global__ void gauss_from_pos(const int* __restrict__ ei, const float* __restrict__ pos,
                               const float* __restrict__ mx,
                               const float* __restrict__ mu, const float* __restrict__ sig,
                               float* __restrict__ gauss, int E)
{
    int e = blockIdx.x * blockDim.x + threadIdx.x;
    if (e >= E) return;
    int r = ei[e], c = ei[E + e];
    float inv = 1.0f / (2.0f * (*mx));
    float p0 = (pos[2 * r]     - pos[2 * c])     * inv + 0.5f;
    float p1 = (pos[2 * r + 1] - pos[2 * c + 1]) * inv + 0.5f;
    gauss3(p0, p1, mu, sig, &gauss[3 * (size_t)e]);
}

// ---------------- edge gather/weight/scatter (mean aggregation) ----------
template <int M, int LOGM>
__global__ void edge_scatter(const int* __restrict__ ei, const float* __restrict__ gauss,
                             const float* __restrict__ y, float* __restrict__ acc,
                             float* __restrict__ cnt, int E)
{
    long idx = blockIdx.x * (long)blockDim.x + threadIdx.x;
    int e = (int)(idx >> LOGM);
    if (e >= E) return;
    int m = (int)idx & (M - 1);
    int r = ei[e], c = ei[E + e];
    float w0 = gauss[3 * (size_t)e];
    float w1 = gauss[3 * (size_t)e + 1];
    float w2 = gauss[3 * (size_t)e + 2];
    const float* yb = y + (size_t)r * (3 * M) + m;
    float v = w0 * yb[0] + w1 * yb[M] + w2 * yb[2 * M];
    atomicAdd(acc + (size_t)c * M + m, v);
    if (m == 0) atomicAdd(cnt + c, 1.0f);
}

// ---------------- finish: elu(acc/cnt + root + bias) ---------------------
__global__ void l1_finish(const float* __restrict__ acc, const float* __restrict__ cnt,
                          const float* __restrict__ x, const float* __restrict__ rt,
                          const float* __restrict__ bias, float* __restrict__ h)
{
    long idx = blockIdx.x * (long)blockDim.x + threadIdx.x;
    if (idx >= (long)N_ * 32) return;
    int n = (int)(idx >> 5), m = (int)idx & 31;
    float v = acc[idx] / fmaxf(cnt[n], 1.0f) + x[n] * rt[m] + bias[m];
    h[idx] = eluf(v);
}

template <int M, int LOGM>
__global__ void gmm_finish(const float* __restrict__ acc, const float* __restrict__ cnt,
                           const float* __restrict__ r, const float* __restrict__ bias,
                           float* __restrict__ out, int Nn)
{
    long idx = blockIdx.x * (long)blockDim.x + threadIdx.x;
    int n = (int)(idx >> LOGM);
    if (n >= Nn) return;
    int m = (int)idx & (M - 1);
    float v = acc[idx] / fmaxf(cnt[n], 1.0f) + r[idx] + bias[m];
    out[idx] = eluf(v);
}

// ---------------- segment max pooling (monotonic-uint atomicMax) ---------
__global__ void fill_u32(unsigned* __restrict__ p, unsigned v, long n)
{
    long idx = blockIdx.x * (long)blockDim.x + threadIdx.x;
    if (idx < n) p[idx] = v;
}

template <int M, int LOGM>
__global__ void pool_max(const int* __restrict__ cluster, const float* __restrict__ h,
                         unsigned* __restrict__ out, int Nn)
{
    long idx = blockIdx.x * (long)blockDim.x + threadIdx.x;
    int n = (int)(idx >> LOGM);
    if (n >= Nn) return;
    int m = (int)idx & (M - 1);
    atomicMax(out + (size_t)cluster[n] * M + m, enc_f32(h[idx]));
}

__global__ void pool_fix(float* __restrict__ p, long n)
{
    long idx = blockIdx.x * (long)blockDim.x + threadIdx.x;
    if (idx >= n) return;
    unsigned u = ((unsigned*)p)[idx];
    p[idx] = (u == ENC_NEGINF) ? 0.0f : dec_f32(u);   // empty cluster -> 0
}

// ---------------- segment mean of positions ------------------------------
__global__ void pos_accum(const int* __restrict__ cluster, const float* __restrict__ pos,
                          float* __restrict__ psum, float* __restrict__ pcnt, int Nn)
{
    int n = blockIdx.x * blockDim.x + threadIdx.x;
    if (n >= Nn) return;
    int cl = cluster[n];
    atomicAdd(&psum[2 * cl],     pos[2 * n]);
    atomicAdd(&psum[2 * cl + 1], pos[2 * n + 1]);
    atomicAdd(&pcnt[cl], 1.0f);
}

__global__ void pos_div(float* __restrict__ psum, const float* __restrict__ pcnt, int Nn)
{
    int i = blockIdx.x * blockDim.x + threadIdx.x;
    if (i >= Nn) return;
    float c = fmaxf(pcnt[i], 1.0f);
    psum[2 * i]     /= c;
    psum[2 * i + 1] /= c;
}

// ---------------- global mean pool + FC head + log_softmax ---------------
__global__ void batch_accum(const int* __restrict__ bat, const float* __restrict__ h,
                            float* __restrict__ gsum, float* __restrict__ gcnt, int Nn)
{
    long idx = blockIdx.x * (long)blockDim.x + threadIdx.x;
    int n = (int)(idx >> 6);
    if (n >= Nn) return;
    int m = (int)idx & 63;
    int b = bat[n];
    atomicAdd(&gsum[(size_t)b * 64 + m], h[idx]);
    if (m == 0) atomicAdd(&gcnt[b], 1.0f);
}

__global__ void head_kernel(const float* __restrict__ gsum, const float* __restrict__ gcnt,
                            const float* __restrict__ fc1w, const float* __restrict__ fc1b,
                            const float* __restrict__ fc2w, const float* __restrict__ fc2b,
                            float* __restrict__ out)
{
    __shared__ float gm[64];
    __shared__ float hid[128];
    __shared__ float lg[10];
    __shared__ float lse;
    int b = blockIdx.x, t = threadIdx.x;
    if (t < 64) gm[t] = gsum[(size_t)b * 64 + t] / fmaxf(gcnt[b], 1.0f);
    __syncthreads();
    float s = fc1b[t];
    for (int k = 0; k < 64; k++) s += gm[k] * fc1w[k * 128 + t];
    hid[t] = eluf(s);
    __syncthreads();
    if (t < 10) {
        float v = fc2b[t];
        for (int k = 0; k < 128; k++) v += hid[k] * fc2w[k * 10 + t];
        lg[t] = v;
    }
    __syncthreads();
    if (t == 0) {
        float mx = lg[0];
        for (int i = 1; i < 10; i++) mx = fmaxf(mx, lg[i]);
        float se = 0.0f;
        for (int i = 0; i < 10; i++) se += __expf(lg[i] - mx);
        lse = mx + logf(se);
    }
    __syncthreads();
    if (t < 10) out[b * 10 + t] = lg[t] - lse;
}

// =========================================================================
#define GRID(n, b) ((int)(((long)(n) + (b) - 1) / (b)))

extern "C" void kernel_launch(void* const* d_in, const int* in_sizes, int n_in,
                              void* d_out, int out_size, void* d_ws, size_t ws_size,
                              hipStream_t stream)
{
    const float* x    = (const float*)d_in[0];
    const float* ea   = (const float*)d_in[1];
    const float* pos  = (const float*)d_in[2];
    const int*   ei   = (const int*)d_in[3];
    const int*   cl1  = (const int*)d_in[4];
    const int*   ei1  = (const int*)d_in[5];
    const int*   cl2  = (const int*)d_in[6];
    const int*   ei2  = (const int*)d_in[7];
    const int*   bat2 = (const int*)d_in[8];
    const float* g1   = (const float*)d_in[9];
    const float* mu1  = (const float*)d_in[10];
    const float* sg1  = (const float*)d_in[11];
    const float* rt1  = (const float*)d_in[12];
    const float* b1   = (const float*)d_in[13];
    const float* g2   = (const float*)d_in[14];
    const float* mu2  = (const float*)d_in[15];
    const float* sg2  = (const float*)d_in[16];
    const float* rt2  = (const float*)d_in[17];
    const float* b2   = (const float*)d_in[18];
    const float* g3   = (const float*)d_in[19];
    const float* mu3  = (const float*)d_in[20];
    const float* sg3  = (const float*)d_in[21];
    const float* rt3  = (const float*)d_in[22];
    const float* b3   = (const float*)d_in[23];
    const float* fc1w = (const float*)d_in[24];
    const float* fc1b = (const float*)d_in[25];
    const float* fc2w = (const float*)d_in[26];
    const float* fc2b = (const float*)d_in[27];
    float* out = (float*)d_out;

    // ---- carve workspace (floats) ----
    float* W = (float*)d_ws;
    size_t o = 0;
    float* y1     = W + o; o += (size_t)N_  * 96;
    float* gauss1 = W + o; o += (size_t)E_  * 3;
    float* acc1   = W + o; o += (size_t)N_  * 32;
    float* cnt1   = W + o; o += (size_t)N_;
    float* h      = W + o; o += (size_t)N_  * 32;
    float* h1     = W + o; o += (size_t)N1_ * 32;
    float* p1sum  = W + o; o += (size_t)N1_ * 2;   // becomes pos1 in place
    float* p1cnt  = W + o; o += (size_t)N1_;
    float* mxA    = W + o; o += 1;
    float* gauss2 = W + o; o += (size_t)E1_ * 3;
    float* y2     = W + o; o += (size_t)N1_ * 192;
    float* r2     = W + o; o += (size_t)N1_ * 64;
    float* acc2   = W + o; o += (size_t)N1_ * 64;
    float* cnt2   = W + o; o += (size_t)N1_;
    float* h1o    = W + o; o += (size_t)N1_ * 64;
    float* h2     = W + o; o += (size_t)N2_ * 64;
    float* p2sum  = W + o; o += (size_t)N2_ * 2;   // becomes pos2 in place
    float* p2cnt  = W + o; o += (size_t)N2_;
    float* mxB    = W + o; o += 1;
    float* gauss3 = W + o; o += (size_t)E2_ * 3;
    float* y3     = W + o; o += (size_t)N2_ * 192;
    float* r3     = W + o; o += (size_t)N2_ * 64;
    float* acc3   = W + o; o += (size_t)N2_ * 64;
    float* cnt3   = W + o; o += (size_t)N2_;
    float* h2o    = W + o; o += (size_t)N2_ * 64;
    float* gsum   = W + o; o += (size_t)B_  * 64;
    float* gcnt   = W + o; o += (size_t)B_;

    // ---- zero accumulators (contiguous regions) ----
    hipMemsetAsync(acc1,  0, ((size_t)N_  * 32 + N_)      * 4, stream);
    hipMemsetAsync(p1sum, 0, ((size_t)N1_ * 2 + N1_ + 1)  * 4, stream);
    hipMemsetAsync(acc2,  0, ((size_t)N1_ * 64 + N1_)     * 4, stream);
    hipMemsetAsync(p2sum, 0, ((size_t)N2_ * 2 + N2_ + 1)  * 4, stream);
    hipMemsetAsync(acc3,  0, ((size_t)N2_ * 64 + N2_)     * 4, stream);
    hipMemsetAsync(gsum,  0, ((size_t)B_  * 64 + B_)      * 4, stream);
    fill_u32<<<GRID((long)N1_ * 32, 256), 256, 0, stream>>>((unsigned*)h1, ENC_NEGINF, (long)N1_ * 32);
    fill_u32<<<GRID((long)N2_ * 64, 256), 256, 0, stream>>>((unsigned*)h2, ENC_NEGINF, (long)N2_ * 64);

    // ---- layer 1 ----
    l1_node<<<GRID((long)N_ * 96, 256), 256, 0, stream>>>(x, g1, y1);
    gauss_from_attr<<<GRID(E_, 256), 256, 0, stream>>>(ea, mu1, sg1, gauss1, E_);
    edge_scatter<32, 5><<<GRID((long)E_ * 32, 256), 256, 0, stream>>>(ei, gauss1, y1, acc1, cnt1, E_);
    l1_finish<<<GRID((long)N_ * 32, 256), 256, 0, stream>>>(acc1, cnt1, x, rt1, b1, h);

    // ---- pool 1 ----
    pool_max<32, 5><<<GRID((long)N_ * 32, 256), 256, 0, stream>>>(cl1, h, (unsigned*)h1, N_);
    pos_accum<<<GRID(N_, 256), 256, 0, stream>>>(cl1, pos, p1sum, p1cnt, N_);
    pool_fix<<<GRID((long)N1_ * 32, 256), 256, 0, stream>>>(h1, (long)N1_ * 32);
    pos_div<<<GRID(N1_, 256), 256, 0, stream>>>(p1sum, p1cnt, N1_);

    // ---- layer 2 ----
    cart_absmax<<<GRID(E1_, 256), 256, 0, stream>>>(ei1, p1sum, mxA, E1_);
    gauss_from_pos<<<GRID(E1_, 256), 256, 0, stream>>>(ei1, p1sum, mxA, mu2, sg2, gauss2, E1_);
    {
        int waves = ((N1_ + 15) / 16) * (192 / 16);
        wmma_gemm<32><<<GRID(waves, 4), 128, 0, stream>>>(h1, g2, y2, N1_, 192);
        waves = ((N1_ + 15) / 16) * (64 / 16);
        wmma_gemm<32><<<GRID(waves, 4), 128, 0, stream>>>(h1, rt2, r2, N1_, 64);
    }
    edge_scatter<64, 6><<<GRID((long)E1_ * 64, 256), 256, 0, stream>>>(ei1, gauss2, y2, acc2, cnt2, E1_);
    gmm_finish<64, 6><<<GRID((long)N1_ * 64, 256), 256, 0, stream>>>(acc2, cnt2, r2, b2, h1o, N1_);

    // ---- pool 2 ----
    pool_max<64, 6><<<GRID((long)N1_ * 64, 256), 256, 0, stream>>>(cl2, h1o, (unsigned*)h2, N1_);
    pos_accum<<<GRID(N1_, 256), 256, 0, stream>>>(cl2, p1sum, p2sum, p2cnt, N1_);
    pool_fix<<<GRID((long)N2_ * 64, 256), 256, 0, stream>>>(h2, (long)N2_ * 64);
    pos_div<<<GRID(N2_, 256), 256, 0, stream>>>(p2sum, p2cnt, N2_);

    // ---- layer 3 ----
    cart_absmax<<<GRID(E2_, 256), 256, 0, stream>>>(ei2, p2sum, mxB, E2_);
    gauss_from_pos<<<GRID(E2_, 256), 256, 0, stream>>>(ei2, p2sum, mxB, mu3, sg3, gauss3, E2_);
    {
        int waves = ((N2_ + 15) / 16) * (192 / 16);
        wmma_gemm<64><<<GRID(waves, 4), 128, 0, stream>>>(h2, g3, y3, N2_, 192);
        waves = ((N2_ + 15) / 16) * (64 / 16);
        wmma_gemm<64><<<GRID(waves, 4), 128, 0, stream>>>(h2, rt3, r3, N2_, 64);
    }
    edge_scatter<64, 6><<<GRID((long)E2_ * 64, 256), 256, 0, stream>>>(ei2, gauss3, y3, acc3, cnt3, E2_);
    gmm_finish<64, 6><<<GRID((long)N2_ * 64, 256), 256, 0, stream>>>(acc3, cnt3, r3, b3, h2o, N2_);

    // ---- head ----
    batch_accum<<<GRID((long)N2_ * 64, 256), 256, 0, stream>>>(bat2, h2o, gsum, gcnt, N2_);
    head_kernel<<<B_, 128, 0, stream>>>(gsum, gcnt, fc1w, fc1b, fc2w, fc2b, out);
}